// Predictor_26714696581172
// MI455X (gfx1250) — compile-verified
//
#include <hip/hip_runtime.h>
#include <hip/hip_bf16.h>

typedef float v2f __attribute__((ext_vector_type(2)));
typedef float v8f __attribute__((ext_vector_type(8)));

#define B_   8
#define L_   384
#define Q_   64
#define H_   256
#define T1_  10
#define P_   73920            // L*(L+1)/2
#define LBL_ 5120             // B*Q*T1

// ---------------------------------------------------------------------------
// Wave-per-(16x64)-tile fp32 WMMA GEMM:  C = alpha * (A x B) + bias
//   A element (m,k) at Ap[m*LDA + k]
//   B element (k,n) at Bp[k*BSK + n*BSN]   (all strides compile-time ->
//                                           immediate offsets, b64 merges)
//   One wave holds 4 accumulators (N-adjacent 16x16 tiles); the A fragment is
//   loaded once per k-step and reused by 4 WMMAs.
// ---------------------------------------------------------------------------
template <int LDA, int BSK, int BSN, int LDC, int K, int N, bool HASBIAS>
__global__ __launch_bounds__(32) void wmma_gemm(
    const float* __restrict__ A, long aY, long aZ,
    const float* __restrict__ Bm, long bY, long bZ,
    float* __restrict__ C, long cY, long cZ,
    const float* __restrict__ bias, float alpha)
{
    constexpr int NT4 = N / 64;
    const int tileN4 = blockIdx.x % NT4;
    const int tileM  = blockIdx.x / NT4;

    const float* Ap = A  + (long)blockIdx.y * aY + (long)blockIdx.z * aZ;
    const float* Bp = Bm + (long)blockIdx.y * bY + (long)blockIdx.z * bZ;
    float*       Cp = C  + (long)blockIdx.y * cY + (long)blockIdx.z * cZ;

    const int lane = threadIdx.x;
    const int lr   = lane & 15;         // row/col within fragment
    const int lh   = lane >> 4;         // half-wave selector
    const int mA   = tileM * 16 + lr;   // A row this lane loads
    const int nB   = tileN4 * 64 + lr;  // base B col this lane loads

    v8f acc0 = {}, acc1 = {}, acc2 = {}, acc3 = {};
    const float* arow = Ap + (long)mA * LDA;

    for (int k = 0; k < K; k += 4) {
        const int ka = k + (lh << 1);   // lanes 0-15: k,k+1 ; 16-31: k+2,k+3
        v2f a;
        a.x = arow[ka];
        a.y = arow[ka + 1];
        if (k + 16 < K) {               // global_prefetch_b8 of next k-slab
            __builtin_prefetch(arow + ka + 16, 0, 1);
            __builtin_prefetch(Bp + (long)(ka + 16) * BSK + (long)nB * BSN, 0, 1);
        }
        #pragma unroll
        for (int j = 0; j < 4; ++j) {
            const int n = nB + j * 16;
            v2f b;
            b.x = Bp[(long)ka * BSK + (long)n * BSN];
            b.y = Bp[(long)(ka + 1) * BSK + (long)n * BSN];
            if (j == 0) acc0 = __builtin_amdgcn_wmma_f32_16x16x4_f32(false, a, false, b, (short)0, acc0, false, false);
            if (j == 1) acc1 = __builtin_amdgcn_wmma_f32_16x16x4_f32(false, a, false, b, (short)0, acc1, false, false);
            if (j == 2) acc2 = __builtin_amdgcn_wmma_f32_16x16x4_f32(false, a, false, b, (short)0, acc2, false, false);
            if (j == 3) acc3 = __builtin_amdgcn_wmma_f32_16x16x4_f32(false, a, false, b, (short)0, acc3, false, false);
        }
    }

    #pragma unroll
    for (int j = 0; j < 4; ++j) {
        const int n  = nB + j * 16;
        const v8f& acc = (j == 0) ? acc0 : (j == 1) ? acc1 : (j == 2) ? acc2 : acc3;
        const float bv = HASBIAS ? bias[n] : 0.0f;
        #pragma unroll
        for (int r = 0; r < 8; ++r) {
            const int m = tileM * 16 + r + (lh << 3);  // VGPR r: M=r (lo)/r+8 (hi)
            Cp[(long)m * LDC + n] = acc[r] * alpha + bv;
        }
    }
}

// ---------------------------------------------------------------------------
// labels_pros = softmax(queries_logits + queries@Wc + bc)  — one wave per (b,q)
// ---------------------------------------------------------------------------
__global__ __launch_bounds__(32) void labels_kernel(
    const float* __restrict__ queries, const float* __restrict__ logits,
    const float* __restrict__ Wc, const float* __restrict__ bc,
    float* __restrict__ out)
{
    const int bq = blockIdx.x, lane = threadIdx.x;
    __shared__ float lg[T1_];
    __shared__ float mx, sum;
    if (lane < T1_) {
        const float* qr = queries + (long)bq * H_;
        float s = bc[lane];
        for (int h = 0; h < H_; ++h) s += qr[h] * Wc[h * T1_ + lane];
        lg[lane] = s + logits[bq * T1_ + lane];
    }
    __syncthreads();
    if (lane == 0) {
        float m = lg[0];
        for (int t = 1; t < T1_; ++t) m = fmaxf(m, lg[t]);
        float s = 0.f;
        for (int t = 0; t < T1_; ++t) s += __expf(lg[t] - m);
        mx = m; sum = s;
    }
    __syncthreads();
    if (lane < T1_) out[bq * T1_ + lane] = __expf(lg[lane] - mx) / sum;
}

// ---------------------------------------------------------------------------
// Per-(b,q) pair-softmax statistics, all O(L):
//   m   = max_e( prefixmax(S0)[e] + S1[e] )                (exact span max)
//   A0[s] = S0[s] - fx*(s-lx)^2
//   A1[e] = S1[e] - fy*(e-ly)^2 - m   (or -inf if masked)
//   sum = sum_e exp(A1[e]) * prefixsum(exp(A0))[e]         (exact denominator)
// ---------------------------------------------------------------------------
__global__ __launch_bounds__(128) void pair_stats_kernel(
    const float* __restrict__ score,         // [B][2][Q][L]
    const unsigned char* __restrict__ masks, // [B][L]
    const float* __restrict__ queries,       // [B][Q][H]
    const float* __restrict__ qlocs,         // [B][Q][2]
    const float* __restrict__ Wo, const float* __restrict__ bo,
    const float* __restrict__ Wf, const float* __restrict__ bf,
    float* __restrict__ Aarr,                // [B*Q][2][L]
    float* __restrict__ invArr)              // [B*Q]
{
    const int bq = blockIdx.x, tid = threadIdx.x;
    const int b = bq / Q_, q = bq % Q_;
    __shared__ float S0[L_], S1[L_], A0[L_], A1[L_];
    __shared__ float params[4];              // lx, ly, fx, fy
    __shared__ float mshare;

    const float* s0p = score + ((long)(b * 2 + 0) * Q_ + q) * L_;
    const float* s1p = score + ((long)(b * 2 + 1) * Q_ + q) * L_;
    const unsigned char* mk = masks + (long)b * L_;

    for (int v = tid; v < L_; v += blockDim.x) {
        const bool ok = mk[v] != 0;
        S0[v] = ok ? s0p[v] : -1.0e8f;
        S1[v] = ok ? s1p[v] : -1.0e8f;
    }
    if (tid < 4) {   // offset (Wo,bo) and softplus factors (Wf,bf)
        const float* qr = queries + (long)bq * H_;
        const int   d  = tid & 1;
        const float* W = (tid < 2) ? Wo : Wf;
        float s = (tid < 2) ? bo[d] : bf[d];
        for (int h = 0; h < H_; ++h) s += qr[h] * W[h * 2 + d];
        params[tid] = (tid < 2) ? (qlocs[bq * 2 + d] + s)
                                : ((s > 20.f) ? s : __logf(1.f + __expf(s)));
    }
    __syncthreads();
    const float lx = params[0], ly = params[1], fx = params[2], fy = params[3];
    for (int v = tid; v < L_; v += blockDim.x) {
        const float ds = (float)v - lx, de = (float)v - ly;
        A0[v] = S0[v] - fx * ds * ds;
        const float a1 = S1[v] - fy * de * de;
        A1[v] = (mk[v] != 0) ? a1 : -1.0e30f;   // weight==0 -> e==0
    }
    __syncthreads();
    if (tid == 0) {
        float pm = -3.0e38f, m = -3.0e38f;
        for (int e = 0; e < L_; ++e) {          // exact max over start<=end pairs
            pm = fmaxf(pm, S0[e]);
            m  = fmaxf(m, pm + S1[e]);
        }
        float mA0 = -3.0e38f;
        for (int s = 0; s < L_; ++s) mA0 = fmaxf(mA0, A0[s]);
        float p = 0.f, sum = 0.f;
        for (int e = 0; e < L_; ++e) {          // factorized denominator
            p   += __expf(A0[e] - mA0);
            sum += __expf(A1[e] - m + mA0) * p;
        }
        mshare = m;
        invArr[bq] = 1.0f / (sum + 1e-12f);
    }
    __syncthreads();
    const float m = mshare;
    float* Ao = Aarr + (long)bq * 2 * L_;
    for (int v = tid; v < L_; v += blockDim.x) {
        Ao[v]      = A0[v];
        Ao[L_ + v] = A1[v] - m;
    }
}

// ---------------------------------------------------------------------------
// Single pass over the 151 MB output: out[bq][tri(e)+s] = exp(A0[s]+A1[e])*inv
// blockIdx.y selects a group of EG_ consecutive `end` values; stores coalesce
// along `s`.
// ---------------------------------------------------------------------------
#define EG_ 48   // L_/EG_ = 8 groups
__global__ __launch_bounds__(256) void pair_write_kernel(
    const float* __restrict__ Aarr, const float* __restrict__ invArr,
    float* __restrict__ out)          // points at locs_pros section
{
    const int bq = blockIdx.x, g = blockIdx.y, tid = threadIdx.x;
    __shared__ float A0s[L_], A1s[EG_];
    const float* Ao = Aarr + (long)bq * 2 * L_;
    for (int v = tid; v < L_; v += blockDim.x) A0s[v] = Ao[v];
    const int e0 = g * EG_;
    if (tid < EG_) A1s[tid] = Ao[L_ + e0 + tid];
    __syncthreads();
    const float inv = invArr[bq];
    float* orow = out + (long)bq * P_;
    for (int e = e0; e < e0 + EG_; ++e) {
        const float a1   = A1s[e - e0];
        const long  base = (long)e * (e + 1) / 2;
        for (int s = tid; s <= e; s += blockDim.x)
            orow[base + s] = __expf(A0s[s] + a1) * inv;
    }
}

// ---------------------------------------------------------------------------
extern "C" void kernel_launch(void* const* d_in, const int* in_sizes, int n_in,
                              void* d_out, int out_size, void* d_ws, size_t ws_size,
                              hipStream_t stream) {
    const float*         hiddens = (const float*)d_in[0];          // B,L,H
    const unsigned char* masks   = (const unsigned char*)d_in[1];  // B,L (bool)
    const float*         queries = (const float*)d_in[2];          // B,Q,H
    const float*         qlocs   = (const float*)d_in[3];          // B,Q,2
    const float*         qlogits = (const float*)d_in[4];          // B,Q,T+1
    const float*         Wc = (const float*)d_in[5];
    const float*         bc = (const float*)d_in[6];
    const float*         Wq = (const float*)d_in[7];
    const float*         bqv = (const float*)d_in[8];
    const float*         Wv = (const float*)d_in[9];
    const float*         bv = (const float*)d_in[10];
    const float*         Wo = (const float*)d_in[11];
    const float*         bo = (const float*)d_in[12];
    const float*         Wf = (const float*)d_in[13];
    const float*         bf = (const float*)d_in[14];
    float* out = (float*)d_out;

    float* ws    = (float*)d_ws;
    float* qh    = ws;                  // [B*Q][2H]      262144
    float* ch    = qh + 262144;         // [B*L][2H]     1572864
    float* score = ch + 1572864;        // [B][2][Q][L]   393216
    float* Aarr  = score + 393216;      // [B*Q][2][L]    393216
    float* invA  = Aarr + 393216;       // [B*Q]             512

    // qh = queries @ Wq + bq     (M=512, N=512, K=256)
    wmma_gemm<H_, 2*H_, 1, 2*H_, H_, 2*H_, true>
        <<<dim3((512/16)*(512/64), 1, 1), 32, 0, stream>>>(
        queries, 0, 0,  Wq, 0, 0,  qh, 0, 0,  bqv, 1.0f);

    // ch = hiddens @ Wv + bv     (M=3072, N=512, K=256)
    wmma_gemm<H_, 2*H_, 1, 2*H_, H_, 2*H_, true>
        <<<dim3((3072/16)*(512/64), 1, 1), 32, 0, stream>>>(
        hiddens, 0, 0,  Wv, 0, 0,  ch, 0, 0,  bv, 1.0f);

    // score[b,c] = (qh[b,:,c,:] @ ch[b,:,c,:]^T) / 16   (M=64, N=384, K=256)
    // B supplied transposed via compile-time strides: B(k,n) = ch[b, n, c, k]
    wmma_gemm<2*H_, 1, 2*H_, L_, H_, L_, false>
        <<<dim3((Q_/16)*(L_/64), 2, B_), 32, 0, stream>>>(
        qh, (long)H_, (long)Q_*2*H_,
        ch, (long)H_, (long)L_*2*H_,
        score, (long)Q_*L_, (long)2*Q_*L_,
        nullptr, 0.0625f);

    labels_kernel<<<dim3(B_*Q_), 32, 0, stream>>>(queries, qlogits, Wc, bc, out);

    pair_stats_kernel<<<dim3(B_*Q_), 128, 0, stream>>>(
        score, masks, queries, qlocs, Wo, bo, Wf, bf, Aarr, invA);

    pair_write_kernel<<<dim3(B_*Q_, L_/EG_), 256, 0, stream>>>(
        Aarr, invA, out + LBL_);
}